// HarmonicNoiseSynth_71201967833406
// MI455X (gfx1250) — compile-verified
//
#include <hip/hip_runtime.h>
#include <hip/hip_bf16.h>
#include <cstdint>
#include <math.h>

// Problem constants (from reference setup)
#define B_CNT   4
#define H_CNT   128
#define NB_CNT  32
#define T_LEN   65536
#define CHUNK   512                    // t-elements per block in main kernel
#define CPR     (T_LEN / CHUNK)        // 128 block-chunks per row
#define CH64    64                     // prefix granularity: one wave32 * 2 elems
#define CPR64   (T_LEN / CH64)         // 1024 wave-chunks per row
#define ROWS    (B_CNT * H_CNT)        // 512 (b,h) rows
#define THREADS 256
#define OMEGA_K 1.3089969389957471e-4f // 2*pi/48000
#define NYQ     24000.0f
#define TWO_OVER_PI 0.63661977236758134f

// ---------------- wave32 helpers ----------------
__device__ __forceinline__ float wave_incl_scan(float v) {
  const int lane = threadIdx.x & 31;
#pragma unroll
  for (int d = 1; d < 32; d <<= 1) {
    float y = __shfl_up(v, d, 32);
    if (lane >= d) v += y;
  }
  return v;
}

__device__ __forceinline__ float wave_reduce(float v) {
#pragma unroll
  for (int d = 16; d > 0; d >>= 1) v += __shfl_down(v, d, 32);
  return v;
}

// ---------------- kernel 1: per-64-element omega sums (one wave per chunk) ----
__global__ void k_chunksum(const float* __restrict__ freq,
                           float* __restrict__ chunkSums) {
  const int blk = blockIdx.x;           // row*(T/512) + seg
  const int row = blk >> 7;
  const int seg = blk & (CPR - 1);
  const int wid = threadIdx.x >> 5;
  const float2 f = *(const float2*)(freq + (size_t)row * T_LEN +
                                    (size_t)seg * CHUNK + (size_t)threadIdx.x * 2);
  const float o0 = (f.x < NYQ ? f.x : 0.0f) * OMEGA_K;
  const float o1 = (f.y < NYQ ? f.y : 0.0f) * OMEGA_K;
  const float v = wave_reduce(o0 + o1);
  if ((threadIdx.x & 31) == 0)
    chunkSums[(size_t)row * CPR64 + (size_t)seg * 8 + wid] = v;
}

// -------- kernel 2: fp64 exclusive scan over 1024 chunks/row (1 wave/row) ----
// Output is TRANSPOSED: chunkBase[c64 * ROWS + row]  (h-contiguous for k_main)
__global__ void k_scanrows(const float* __restrict__ chunkSums,
                           const float* __restrict__ init_phase,
                           float* __restrict__ chunkBase) {
  const int lane = threadIdx.x & 31;
  const int wid  = threadIdx.x >> 5;
  const int r    = blockIdx.x * 8 + wid;          // 64 blocks * 8 waves = 512 rows
  const double TWO_PI = 6.283185307179586476925286766559;
  const float* src = chunkSums + (size_t)r * CPR64;

  // lane owns 32 consecutive chunks
  double lsum = 0.0;
  for (int i = 0; i < 32; ++i) lsum += (double)src[lane * 32 + i];
  double incl = lsum;
#pragma unroll
  for (int d = 1; d < 32; d <<= 1) {
    double y = __shfl_up(incl, d, 32);
    if (lane >= d) incl += y;
  }
  double acc = incl - lsum;                       // exclusive prefix
  const double ip = (double)init_phase[r];
  for (int i = 0; i < 32; ++i) {
    const int c = lane * 32 + i;
    double base = acc + ip;
    chunkBase[(size_t)c * ROWS + r] = (float)(base - floor(base / TWO_PI) * TWO_PI);
    acc += (double)src[c];
  }
}

// ---------------- kernel 3: main synth (barrier-free, 4-deep async pipe) -----
__global__ void k_main(const float* __restrict__ freq,
                       const float* __restrict__ amp,
                       const float* __restrict__ nbamp,
                       const float* __restrict__ mexp,
                       const float* __restrict__ mw,
                       const float* __restrict__ pulse_g,
                       const float* __restrict__ flow_g,
                       const float* __restrict__ bands,
                       const float* __restrict__ chunkBase,
                       float* __restrict__ out) {
  __shared__ __align__(16) float fbuf[4][CHUNK];
  __shared__ __align__(16) float abuf[4][CHUNK];
  __shared__ float basebuf[8][H_CNT];   // per-wave staged phase bases

  const int tid  = threadIdx.x;
  const int lane = tid & 31;
  const int wid  = tid >> 5;
  const int blk  = blockIdx.x;          // b*CPR + chunk
  const int b    = blk >> 7;
  const int ch   = blk & (CPR - 1);
  const size_t elemOff = (size_t)ch * CHUNK + (size_t)tid * 2;

  // modulator constants (avoid in-loop global loads behind asm clobbers)
  const float me0 = mexp[b * 4 + 0], me1 = mexp[b * 4 + 1];
  const float me2 = mexp[b * 4 + 2], me3 = mexp[b * 4 + 3];
  const float mw0 = mw[b * 4 + 0], mw1 = mw[b * 4 + 1];
  const float mw2 = mw[b * 4 + 2], mw3 = mw[b * 4 + 3];

  // stage this wave's 128 per-h phase bases into LDS (coalesced; own-wave only,
  // DS ops are in-order per wave -> no barrier needed)
  {
    const int c64 = ch * 8 + wid;
    const float* cbp = chunkBase + (size_t)c64 * ROWS + (size_t)b * H_CNT;
#pragma unroll
    for (int q = 0; q < 4; ++q)
      basebuf[wid][q * 32 + lane] = cbp[q * 32 + lane];
  }

  // stage one h-slice of freq+amp (8 bytes/lane each) into LDS via async copy
  auto issue = [&](int hh, int slot) {
    const float* gf = freq + ((size_t)(b * H_CNT + hh)) * T_LEN + elemOff;
    const float* ga = amp  + ((size_t)(b * H_CNT + hh)) * T_LEN + elemOff;
    unsigned fo = (unsigned)(uintptr_t)(&fbuf[slot][tid * 2]);
    unsigned ao = (unsigned)(uintptr_t)(&abuf[slot][tid * 2]);
    asm volatile("global_load_async_to_lds_b64 %0, %1, off"
                 :: "v"(fo), "v"((unsigned long long)(uintptr_t)gf) : "memory");
    asm volatile("global_load_async_to_lds_b64 %0, %1, off"
                 :: "v"(ao), "v"((unsigned long long)(uintptr_t)ga) : "memory");
  };

  issue(0, 0); issue(1, 1); issue(2, 2);          // prime 3 slices

  float accH0 = 0.0f, accH1 = 0.0f;               // harmonic component
  float accM0 = 0.0f, accM1 = 0.0f;               // modulator (waveshaped) sum

  for (int h = 0; h < H_CNT; ++h) {
    const int hn = (h + 3 < H_CNT) ? (h + 3) : (H_CNT - 1);
    issue(hn, (h + 3) & 3);                       // prefetch 3 ahead
    asm volatile("s_wait_asynccnt 0x6" ::: "memory");  // this h's pair is done

    const int cur = h & 3;
    const float2 fv = *(const float2*)&fbuf[cur][tid * 2];
    const float2 av = *(const float2*)&abuf[cur][tid * 2];

    const float o0 = (fv.x < NYQ ? fv.x : 0.0f) * OMEGA_K;
    const float o1 = (fv.y < NYQ ? fv.y : 0.0f) * OMEGA_K;
    const float local = o0 + o1;

    // per-wave exclusive prefix along this wave's 64-sample strip (no barriers)
    const float incl  = wave_incl_scan(local);
    const float pexcl = incl - local;

    const float base = basebuf[wid][h];           // cheap LDS read
    const float ph0 = base + pexcl + o0;
    const float ph1 = ph0 + o1;
    const float c0 = __cosf(ph0);
    const float c1 = __cosf(ph1);

    accH0 = fmaf(av.x, c0, accH0);
    accH1 = fmaf(av.y, c1, accH1);

    if (h < 4) {                                  // modulator path
      const float e = (h == 0) ? me0 : (h == 1) ? me1 : (h == 2) ? me2 : me3;
      const float w = (h == 0) ? mw0 : (h == 1) ? mw1 : (h == 2) ? mw2 : mw3;
      const float ms0 = asinf(c0 * 0.99f) * TWO_OVER_PI;
      const float ms1 = asinf(c1 * 0.99f) * TWO_OVER_PI;
      accM0 = fmaf(__powf(fabsf(ms0), e), w, accM0);
      accM1 = fmaf(__powf(fabsf(ms1), e), w, accM1);
    }
  }

  // noise = sum_nb nbamp[b,nb,t] * bands[nb,t]
  float n0 = 0.0f, n1 = 0.0f;
  const float* na = nbamp + (size_t)b * NB_CNT * T_LEN + elemOff;
  const float* bd = bands + elemOff;
  for (int nb = 0; nb < NB_CNT; ++nb) {
    if (nb + 2 < NB_CNT) {
      __builtin_prefetch(na + (size_t)(nb + 2) * T_LEN, 0, 1);
      __builtin_prefetch(bd + (size_t)(nb + 2) * T_LEN, 0, 1);
    }
    const float2 avn = *(const float2*)(na + (size_t)nb * T_LEN);
    const float2 bvn = *(const float2*)(bd + (size_t)nb * T_LEN);
    n0 = fmaf(avn.x, bvn.x, n0);
    n1 = fmaf(avn.y, bvn.y, n1);
  }

  const float pg = pulse_g[b];
  const float fg = flow_g[b];
  const float tg = (pg + fg) * 0.7f;

  float2 o;
  o.x = accH0 + accM0 * n0 * pg + accH0 * n0 * tg + n0 * fg * 0.3f;
  o.y = accH1 + accM1 * n1 * pg + accH1 * n1 * tg + n1 * fg * 0.3f;
  *(float2*)(out + (size_t)b * T_LEN + elemOff) = o;
}

// ---------------- launcher ----------------
extern "C" void kernel_launch(void* const* d_in, const int* in_sizes, int n_in,
                              void* d_out, int out_size, void* d_ws, size_t ws_size,
                              hipStream_t stream) {
  (void)in_sizes; (void)n_in; (void)out_size; (void)ws_size;
  const float* freq   = (const float*)d_in[0];  // (B,H,T)
  const float* amp    = (const float*)d_in[1];  // (B,H,T)
  const float* nbamp  = (const float*)d_in[2];  // (B,NB,T)
  const float* mexp   = (const float*)d_in[3];  // (B,4,1)
  const float* mw     = (const float*)d_in[4];  // (B,4,1)
  const float* pg     = (const float*)d_in[5];  // (B,1,1)
  const float* fg     = (const float*)d_in[6];  // (B,1,1)
  const float* iph    = (const float*)d_in[7];  // (B,H,1)
  const float* bands  = (const float*)d_in[8];  // (NB,T)
  float* out = (float*)d_out;

  float* chunkSums = (float*)d_ws;                        // ROWS*CPR64 floats (2 MB)
  float* chunkBase = chunkSums + (size_t)ROWS * CPR64;    // ROWS*CPR64 floats (2 MB), transposed

  k_chunksum<<<ROWS * CPR, THREADS, 0, stream>>>(freq, chunkSums);
  k_scanrows<<<ROWS / 8, THREADS, 0, stream>>>(chunkSums, iph, chunkBase);
  k_main<<<B_CNT * CPR, THREADS, 0, stream>>>(freq, amp, nbamp, mexp, mw, pg, fg,
                                              bands, chunkBase, out);
}